// MultiHeadHash_85985245266455
// MI455X (gfx1250) — compile-verified
//
#include <hip/hip_runtime.h>
#include <cstdint>

// ---------------------------------------------------------------------------
// MultiHeadHash on MI455X (gfx1250)
//
// Bandwidth-bound streaming kernel (~49 MiB total traffic -> ~2.1us floor at
// 23.3 TB/s). All math reduced to u32 (mod 2^20 + XOR => only low 32 bits of
// each product matter). Input ids are staged into LDS with the CDNA5 Tensor
// Data Mover (tensor_load_to_lds, TENSORcnt), which is the gfx1250-specific
// data-movement path appropriate for this kernel; WMMA is inapplicable
// (XOR-reduction is not a matmul).
//
// Toolchain note: this build uses the 6-arg tensor_load_to_lds builtin
// (amdgpu-toolchain / clang-23 form): (v4u, v8i, v4i, v4i, v8i, i32 cpol).
// ---------------------------------------------------------------------------

typedef __attribute__((ext_vector_type(4))) unsigned int tdm_v4u;
typedef __attribute__((ext_vector_type(8))) int          tdm_v8i;
typedef __attribute__((ext_vector_type(4))) int          tdm_v4i;

#define POS_PER_BLOCK 256
#define BYTES_PER_POS 32u                    // 4 ngram ids x int64
#define TILE_BYTES    (POS_PER_BLOCK * 32)   // 8 KB per block
#define TILE_DWORDS   (TILE_BYTES / 4)       // 2048 dwords

__global__ __launch_bounds__(POS_PER_BLOCK)
void MultiHeadHash_85985245266455_kernel(const unsigned long long* __restrict__ ids,
                                         const unsigned char*      __restrict__ msk,
                                         const unsigned long long* __restrict__ pw,
                                         const int*                __restrict__ tsz,
                                         float*                    __restrict__ out,
                                         unsigned                  n_pos)
{
    __shared__ unsigned char tile[TILE_BYTES];

    const unsigned tid       = threadIdx.x;
    const unsigned blockPos0 = blockIdx.x * POS_PER_BLOCK;

    // ---- TDM: DMA this block's 8 KB slab of ngram_ids into LDS (wave 0) ----
    const unsigned wave = (unsigned)__builtin_amdgcn_readfirstlane((int)(tid >> 5));
    if (wave == 0) {
        const unsigned lds_off = (unsigned)(uintptr_t)(&tile[0]);   // low 32 bits = LDS byte offset
        const unsigned long long ga =
            (unsigned long long)(uintptr_t)ids +
            (unsigned long long)blockPos0 * BYTES_PER_POS;

        // D# group 0: count=1, lds_addr, global_addr[56:0], type=2 ("image")
        tdm_v4u g0 = {
            1u,
            lds_off,
            (unsigned)(ga & 0xFFFFFFFFull),
            (unsigned)((ga >> 32) & 0x01FFFFFFull) | (2u << 30)
        };
        // D# group 1: data_size=2 (4B); tensor_dim0 = tile_dim0 = 2048 dwords (1-D tile)
        tdm_v8i g1 = {
            (int)(2u << 16),                              // [17:16] data_size = 4B
            (int)((TILE_DWORDS & 0xFFFFu) << 16),         // [63:48] tensor_dim0 lo16
            (int)(TILE_DWORDS >> 16),                     // [79:64] tensor_dim0 hi16 (dim1=0)
            (int)((TILE_DWORDS & 0xFFFFu) << 16),         // [127:112] tile_dim0
            0, 0, 0, 0
        };
        tdm_v4i gz4 = {0, 0, 0, 0};
        tdm_v8i gz8 = {0, 0, 0, 0, 0, 0, 0, 0};
        __builtin_amdgcn_tensor_load_to_lds(g0, g1, gz4, gz4, gz8, 0);
        __builtin_amdgcn_s_wait_tensorcnt(0);             // per-wave TENSORcnt == 0
    }
    asm volatile("" ::: "memory");   // keep LDS reads after the DMA completion
    __syncthreads();                 // publish LDS tile to all 8 waves

    // ---- per-thread hash: one position, 16 primes, all in u32 -------------
    const unsigned pos = blockPos0 + tid;

    const uint4* lp = reinterpret_cast<const uint4*>(tile) + (tid << 1);
    uint4 lo = lp[0];                // ids 0,1 (int64 -> low dwords at .x/.z)
    uint4 hi = lp[1];                // ids 2,3
    const unsigned id0 = lo.x, id1 = lo.z, id2 = hi.x, id3 = hi.z;

    const unsigned m     = (pos < n_pos && msk[pos]) ? 1u : 0u;
    const unsigned tmask = (unsigned)tsz[0] - 1u;    // table_size is 2^20 (pow2)

    float r[16];
#pragma unroll
    for (int p = 0; p < 16; ++p) {
        // prime_powers row p: [1, p, p^2, ..., p^7] as int64; low dwords suffice
        const unsigned P1 = (unsigned)pw[p * 8 + 1];
        const unsigned P2 = (unsigned)pw[p * 8 + 2];
        const unsigned P3 = (unsigned)pw[p * 8 + 3];
        unsigned h = id0 ^ (id1 * P1) ^ (id2 * P2) ^ (id3 * P3);
        h = (h & tmask) * m;
        r[p] = (float)h;             // < 2^20: exact in float32
    }

    if (pos < n_pos) {
        float4* op = reinterpret_cast<float4*>(out + (size_t)pos * 16u);
        op[0] = make_float4(r[0],  r[1],  r[2],  r[3]);
        op[1] = make_float4(r[4],  r[5],  r[6],  r[7]);
        op[2] = make_float4(r[8],  r[9],  r[10], r[11]);
        op[3] = make_float4(r[12], r[13], r[14], r[15]);
    }
}

extern "C" void kernel_launch(void* const* d_in, const int* in_sizes, int n_in,
                              void* d_out, int out_size, void* d_ws, size_t ws_size,
                              hipStream_t stream)
{
    const unsigned long long* ids = (const unsigned long long*)d_in[0]; // int64 ids
    const unsigned char*      msk = (const unsigned char*)d_in[1];      // bool mask
    const unsigned long long* pw  = (const unsigned long long*)d_in[2]; // int64 16x8
    const int*                tsz = (const int*)d_in[3];                // table_size
    float*                    out = (float*)d_out;

    const unsigned n_pos  = (unsigned)in_sizes[1];                      // batch*seq
    const unsigned blocks = (n_pos + POS_PER_BLOCK - 1) / POS_PER_BLOCK;

    hipLaunchKernelGGL(MultiHeadHash_85985245266455_kernel,
                       dim3(blocks), dim3(POS_PER_BLOCK), 0, stream,
                       ids, msk, pw, tsz, out, n_pos);
}